// GCN_9259949490770
// MI455X (gfx1250) — compile-verified
//
#include <hip/hip_runtime.h>
#include <hip/hip_bf16.h>

// ---------------------------------------------------------------------------
// GCN (2-layer, PyG GCNConv semantics) for MI455X / gfx1250.
//   N=50000 nodes, D=128 feats, E=800000 edges, PRESERVE=0.1
//
// - Dense transform h = x @ W uses V_WMMA_F32_16X16X4_F32 (fp32, matching the
//   fp32 reference). A-tile (16x128, 8KB) is staged in LDS once per block and
//   shared by all 8 waves (ds_load_b64 fragments); W is read once per block
//   straight from the 192MB L2.
// - Self-loop term + bias (agg = h*dinv^2 + b) is fused into the GEMM
//   epilogue, killing a full 51MB read+write pass over the feature matrix.
// - Edge scatter: 1 wave/edge, float4 (b128) gather of h[src], 4 f32 global
//   atomics/lane resolving in L2 (the 25.6MB accumulator is L2-resident).
// - Degree normalization depends only on the graph -> computed once.
// ---------------------------------------------------------------------------

#define GCN_N 50000
#define GCN_D 128
#define GCN_E 800000
#define GCN_PRESERVE 0.1f

typedef __attribute__((ext_vector_type(2))) float v2f;
typedef __attribute__((ext_vector_type(8))) float v8f;

static __host__ inline int cdiv(int a, int b) { return (a + b - 1) / b; }

// ---------------------------------------------------------------- degrees --
__global__ void deg_init_kernel(float* __restrict__ deg, int n) {
    int i = blockIdx.x * blockDim.x + threadIdx.x;
    if (i < n) deg[i] = 1.0f;  // self-loop weight
}

__global__ void deg_accum_kernel(const int* __restrict__ dst,
                                 const float* __restrict__ ew,
                                 float* __restrict__ deg, int e) {
    int i = blockIdx.x * blockDim.x + threadIdx.x;
    if (i < e) atomicAdd(&deg[dst[i]], ew[i]);
}

__global__ void deg_to_dinv_kernel(float* __restrict__ deg, int n) {
    int i = blockIdx.x * blockDim.x + threadIdx.x;
    if (i < n) {
        float d = deg[i];
        deg[i] = (d > 0.0f) ? rsqrtf(d) : 0.0f;  // v_rsq_f32
    }
}

// ---------------------------------------- WMMA GEMM + fused self-loop/bias --
// h[N,128] = x[N,128] @ W[128,128]; agg[N,128] = h * dinv^2 + bias.
// One block = 8 waves; wave w computes the 16x16 tile at (blockIdx.x*16, w*16).
// A frag (16x4): lanes 0-15 hold M=lane, K={k,k+1}; lanes 16-31 hold K={k+2,k+3}
//   -> contiguous pair -> one ds_load_b64 per step from the LDS-staged tile.
// B frag (4x16): lanes 0-15 hold N=lane with K={k,k+1}; lanes 16-31 K={k+2,k+3}.
// C/D: VGPR r holds (M = r + 8*(lane>=16), N = lane&15).
__global__ __launch_bounds__(256)
void gemm_wmma_f32_kernel(const float* __restrict__ x,
                          const float* __restrict__ W,
                          const float* __restrict__ dinv,
                          const float* __restrict__ bias,
                          float* __restrict__ h,
                          float* __restrict__ agg) {
    __shared__ float As[16 * GCN_D];  // 8KB A-tile, shared by all 8 waves

    const int tid = threadIdx.x;
    const int m0  = blockIdx.x * 16;  // 50000 = 3125*16, no remainder

    // cooperative stage: 2048 floats = 512 float4 = 256 threads x 2
    {
        const float4* s4 = (const float4*)(x + (size_t)m0 * GCN_D);
        float4* d4 = (float4*)As;
        d4[tid]       = s4[tid];
        d4[tid + 256] = s4[tid + 256];
    }
    __syncthreads();

    const int wave  = tid >> 5;   // 0..7 -> N tile
    const int lane  = tid & 31;
    const int lhalf = lane >> 4;  // 0: K pair {k,k+1}, 1: {k+2,k+3}
    const int l16   = lane & 15;
    const int n0    = wave * 16;

    const float* arow = As + l16 * GCN_D;
    v8f acc = {};
#pragma unroll 4
    for (int k = 0; k < GCN_D; k += 4) {
        const int ka = k + lhalf * 2;
        v2f a = *(const v2f*)(arow + ka);                       // ds_load_b64
        v2f bf;
        bf.x = W[(size_t)ka * GCN_D + n0 + l16];
        bf.y = W[(size_t)(ka + 1) * GCN_D + n0 + l16];
        acc = __builtin_amdgcn_wmma_f32_16x16x4_f32(
            /*neg_a=*/false, a, /*neg_b=*/false, bf,
            /*c_mod=*/(short)0, acc, /*reuse_a=*/false, /*reuse_b=*/false);
    }

    // fused epilogue: write h, and seed agg with self-loop + bias
    const int col = n0 + l16;
    const float bcol = bias[col];
#pragma unroll
    for (int r = 0; r < 8; ++r) {
        const int row = m0 + 8 * lhalf + r;
        const float di = dinv[row];
        const float v  = acc[r];
        h[(size_t)row * GCN_D + col]   = v;
        agg[(size_t)row * GCN_D + col] = v * di * di + bcol;
    }
}

// ------------------------------------------------------- edge scatter-add --
// One wave per edge: each lane handles 4 features (128 = 32 lanes * 4).
__global__ __launch_bounds__(256)
void edge_agg_kernel(const int* __restrict__ src,
                     const int* __restrict__ dst,
                     const float* __restrict__ ew,
                     const float* __restrict__ dinv,
                     const float* __restrict__ h,
                     float* __restrict__ agg, int e) {
    int g = blockIdx.x * blockDim.x + threadIdx.x;
    int edge = g >> 5;
    if (edge >= e) return;
    int lane = g & 31;
    int s = src[edge];
    int d = dst[edge];
    float c = dinv[s] * ew[edge] * dinv[d];
    float4 hv = *(const float4*)(h + (size_t)s * GCN_D + lane * 4);  // b128 gather
    float* o = agg + (size_t)d * GCN_D + lane * 4;
    atomicAdd(o + 0, hv.x * c);
    atomicAdd(o + 1, hv.y * c);
    atomicAdd(o + 2, hv.z * c);
    atomicAdd(o + 3, hv.w * c);
}

// ---------------------------------------------------------------- residual -
// out = (1-PRESERVE)*agg + PRESERVE*xin   (out may alias xin: pure elementwise)
__global__ void residual_kernel(const float* __restrict__ agg,
                                const float* __restrict__ xin,
                                float* __restrict__ out, int n4) {
    int g = blockIdx.x * blockDim.x + threadIdx.x;
    if (g >= n4) return;
    float4 a = *(const float4*)(agg + (size_t)g * 4);
    float4 x = *(const float4*)(xin + (size_t)g * 4);
    float4 o;
    o.x = (1.0f - GCN_PRESERVE) * a.x + GCN_PRESERVE * x.x;
    o.y = (1.0f - GCN_PRESERVE) * a.y + GCN_PRESERVE * x.y;
    o.z = (1.0f - GCN_PRESERVE) * a.z + GCN_PRESERVE * x.z;
    o.w = (1.0f - GCN_PRESERVE) * a.w + GCN_PRESERVE * x.w;
    *(float4*)(out + (size_t)g * 4) = o;
}

// ---------------------------------------------------------------------------
extern "C" void kernel_launch(void* const* d_in, const int* in_sizes, int n_in,
                              void* d_out, int out_size, void* d_ws, size_t ws_size,
                              hipStream_t stream) {
    (void)in_sizes; (void)n_in; (void)out_size; (void)ws_size;

    const float* skill_embed = (const float*)d_in[0];          // [N, D]
    const float* Wall        = (const float*)d_in[1];          // [L, D, D]
    const float* ball        = (const float*)d_in[2];          // [L, D]
    const float* edge_attr   = (const float*)d_in[3];          // [E]
    const int*   adj         = (const int*)d_in[4];            // [2, E]
    const int* src = adj;
    const int* dst = adj + GCN_E;
    float* out = (float*)d_out;                                // [N, D]

    // workspace layout (floats): dinv[N] | h[N*D] | agg[N*D]   (~51.4 MB)
    float* dinv = (float*)d_ws;
    float* h    = dinv + ((GCN_N + 255) & ~255);
    float* agg  = h + (size_t)GCN_N * GCN_D;

    const int ND4 = GCN_N * (GCN_D / 4);                       // 1.6M float4s

    // ---- degree normalization (graph-only; once for both layers) ----
    deg_init_kernel<<<cdiv(GCN_N, 256), 256, 0, stream>>>(dinv, GCN_N);
    deg_accum_kernel<<<cdiv(GCN_E, 256), 256, 0, stream>>>(dst, edge_attr, dinv, GCN_E);
    deg_to_dinv_kernel<<<cdiv(GCN_N, 256), 256, 0, stream>>>(dinv, GCN_N);

    // ---- layer 0: x = skill_embed, residual -> d_out ----
    gemm_wmma_f32_kernel<<<GCN_N / 16, 256, 0, stream>>>(skill_embed, Wall,
                                                         dinv, ball, h, agg);
    edge_agg_kernel<<<cdiv(GCN_E * 32, 256), 256, 0, stream>>>(src, dst, edge_attr,
                                                               dinv, h, agg, GCN_E);
    residual_kernel<<<cdiv(ND4, 256), 256, 0, stream>>>(agg, skill_embed, out, ND4);

    // ---- layer 1: x = d_out, residual -> d_out (elementwise, aliasing OK) ----
    gemm_wmma_f32_kernel<<<GCN_N / 16, 256, 0, stream>>>(out, Wall + GCN_D * GCN_D,
                                                         dinv, ball + GCN_D, h, agg);
    edge_agg_kernel<<<cdiv(GCN_E * 32, 256), 256, 0, stream>>>(src, dst, edge_attr,
                                                               dinv, h, agg, GCN_E);
    residual_kernel<<<cdiv(ND4, 256), 256, 0, stream>>>(agg, out, out, ND4);
}